// HGT_30906584662470
// MI455X (gfx1250) — compile-verified
//
#include <hip/hip_runtime.h>
#include <math.h>

typedef __attribute__((ext_vector_type(2))) float v2f;
typedef __attribute__((ext_vector_type(8))) float v8f;

#define H_ 8
#define D_ 16
#define HID_ 128
#define NM_ 100000
#define ND_ 20000
#define NA_ 50000
#define E_ 300000
#define NCLS_ 8
#define ENC_NEG_INF 0x007FFFFFu /* fenc(-inf) */

__device__ __forceinline__ float sigmoidf_(float x) { return 1.f / (1.f + __expf(-x)); }
__device__ __forceinline__ float gelu_exact_(float x) {
    return 0.5f * x * (1.f + erff(x * 0.70710678118654752f));
}
// monotonic unsigned encoding so atomicMax(u32) == float max
__device__ __forceinline__ unsigned fenc_(float f) {
    unsigned u = __float_as_uint(f);
    return (u & 0x80000000u) ? ~u : (u | 0x80000000u);
}
__device__ __forceinline__ float fdec_(unsigned k) {
    return (k & 0x80000000u) ? __uint_as_float(k ^ 0x80000000u) : __uint_as_float(~k);
}

__global__ __launch_bounds__(256) void fill_u32_kernel(unsigned* p, unsigned v, int n) {
    int i = blockIdx.x * blockDim.x + threadIdx.x;
    if (i < n) p[i] = v;
}

// zero-pad Wlin [128,8] -> [128,16] and blin [8] -> [16] so the head GEMM has
// no load guards in its inner loop.
__global__ __launch_bounds__(256) void pad_head_kernel(
    const float* __restrict__ Wlin, const float* __restrict__ blin,
    float* __restrict__ Wpad, float* __restrict__ bpad) {
    int i = blockIdx.x * blockDim.x + threadIdx.x;
    if (i < HID_ * 16) {
        int row = i >> 4, col = i & 15;
        Wpad[i] = (col < NCLS_) ? Wlin[row * NCLS_ + col] : 0.f;
    }
    if (i < 16) bpad[i] = (i < NCLS_) ? blin[i] : 0.f;
}

// ---------------------------------------------------------------------------
// Wave-level fp32 WMMA GEMM: C[M, nout] = op(A)[M,K] @ B[K, NT*16] (+bias)
// K and N = NT*16 are compile-time -> zero guards in the inner loop.
// B is staged into LDS per block (KC=min(K,128) rows at a time, 64 KB max),
// rearranged so each lane's per-step operand is one ds_load_b64:
//   Bs[(k0/4)*2N + half*N + col] = { B[k0+2*half][col], B[k0+2*half+1][col] }
// A operand (V_WMMA_F32_16X16X4_F32): lanes 0-15 K={k0,k0+1}, lanes 16-31
// K={k0+2,k0+3}, row = lane&15. C: VGPR j = row j / j+8 per lane half.
// AGG_A: A elem = gelu(num[row,col]/max(denom[row,col/16],1e-16))
// GATE : C = g*(acc+bias) + (1-g)*xs[row,col], g = sigmoid(*skip)
// ---------------------------------------------------------------------------
template <int K, int NT, bool BIAS, bool AGG_A, bool GATE>
__global__ __launch_bounds__(256) void gemm_wmma_kernel(
    const float* __restrict__ A, const float* __restrict__ B,
    const float* __restrict__ bias, const float* __restrict__ denom,
    const float* __restrict__ xs, const float* __restrict__ skip,
    float* __restrict__ C, int M, int nout, int ldc) {
    constexpr int N  = NT * 16;
    constexpr int KC = (K > 128) ? 128 : K;       // LDS chunk depth (<= 64 KB)
    extern __shared__ float2 Bs[];                // KC*N/2 float2

    const int tid  = threadIdx.x;
    const int wave = tid >> 5;
    const int lane = tid & 31;
    const int half = lane >> 4;
    const int r    = lane & 15;
    const int row0 = (blockIdx.x * 8 + wave) * 16;
    const int arow = min(row0 + r, M - 1);        // clamp: no early return (barriers)
    const float* Arow = A + (size_t)arow * K;

    v8f acc[NT];
#pragma unroll
    for (int t = 0; t < NT; ++t) acc[t] = (v8f)0.f;

    for (int kb = 0; kb < K; kb += KC) {
        __syncthreads();  // protect Bs before refill (2nd chunk)
        for (int i = tid; i < KC * N / 2; i += 256) {
            const int col = i % N;
            const int k   = kb + (i / (2 * N)) * 4 + ((i / N) & 1) * 2;
            float2 bp;
            bp.x = B[(size_t)k * N + col];
            bp.y = B[(size_t)(k + 1) * N + col];
            Bs[i] = bp;
        }
        __syncthreads();
        if (kb + KC < K) __builtin_prefetch(Arow + kb + KC, 0, 0);

#pragma unroll 2
        for (int k0 = 0; k0 < KC; k0 += 4) {
            const int kg = kb + k0 + 2 * half;    // 8B-aligned (kg even)
            float2 ap = *(const float2*)(Arow + kg);
            if (AGG_A) {  // kg, kg+1 share one 16-wide head bucket (k0 % 4 == 0)
                const float dn = fmaxf(denom[arow * H_ + (kg >> 4)], 1e-16f);
                ap.x = gelu_exact_(ap.x / dn);
                ap.y = gelu_exact_(ap.y / dn);
            }
            v2f a; a.x = ap.x; a.y = ap.y;
            float2 bv[NT];
            const int bbase = (k0 >> 2) * (2 * N) + half * N + r;
#pragma unroll
            for (int t = 0; t < NT; ++t) bv[t] = Bs[bbase + t * 16];
#pragma unroll
            for (int t = 0; t < NT; ++t) {
                v2f b; b.x = bv[t].x; b.y = bv[t].y;
                acc[t] = __builtin_amdgcn_wmma_f32_16x16x4_f32(
                    false, a, false, b, (short)0, acc[t], false, false);
            }
        }
    }

    float g = 1.f, gi = 0.f;
    if (GATE) { g = sigmoidf_(skip[0]); gi = 1.f - g; }
#pragma unroll
    for (int t = 0; t < NT; ++t) {
        const int col = t * 16 + r;
        if (col >= nout) continue;
#pragma unroll
        for (int j = 0; j < 8; ++j) {
            const int row = row0 + j + 8 * half;
            if (row >= M) continue;
            float val = acc[t][j];
            if (BIAS) val += bias[col];
            if (GATE) val = g * val + gi * xs[(size_t)row * ldc + col];
            C[(size_t)row * ldc + col] = val;
        }
    }
}

// ---------------------------------------------------------------------------
// Edge pass A: alpha[e,h] = p[h]/sqrt(D) * q[dst,h,:] . (k[src,h,:] @ a_rel[h])
// and segment max over dst via encoded atomicMax.
// ---------------------------------------------------------------------------
__global__ __launch_bounds__(256) void edge_alpha_kernel(
    const int* __restrict__ src, const int* __restrict__ dst,
    const float* __restrict__ Ksrc, const float* __restrict__ Qdst,
    const float* __restrict__ a_rel, const float* __restrict__ p_rel,
    float* __restrict__ alpha, unsigned* __restrict__ segmax, int E) {
    int idx = blockIdx.x * blockDim.x + threadIdx.x;
    if (idx >= E * H_) return;
    const int e = idx >> 3, h = idx & 7;
    const int s = src[e], d = dst[e];
    const float* kv = Ksrc + ((size_t)s * H_ + h) * D_;
    const float* qv = Qdst + ((size_t)d * H_ + h) * D_;
    const float* ar = a_rel + (size_t)h * D_ * D_;
    float ke[D_];
#pragma unroll
    for (int f = 0; f < D_; ++f) ke[f] = 0.f;
#pragma unroll
    for (int dd = 0; dd < D_; ++dd) {
        const float kd = kv[dd];
#pragma unroll
        for (int f = 0; f < D_; ++f) ke[f] += kd * ar[dd * D_ + f];
    }
    float al = 0.f;
#pragma unroll
    for (int f = 0; f < D_; ++f) al += qv[f] * ke[f];
    al *= p_rel[h] * 0.25f;  // 1/sqrt(16)
    alpha[idx] = al;
    atomicMax(&segmax[d * H_ + h], fenc_(al));
}

// ---------------------------------------------------------------------------
// Edge pass B: w = exp(alpha - m[dst]); denom += w; num += w * (v[src]@m_rel)
// ---------------------------------------------------------------------------
__global__ __launch_bounds__(256) void edge_accum_kernel(
    const int* __restrict__ src, const int* __restrict__ dst,
    const float* __restrict__ Vsrc, const float* __restrict__ m_rel,
    const float* __restrict__ alpha, const unsigned* __restrict__ segmax,
    float* __restrict__ denom, float* __restrict__ num, int E) {
    int idx = blockIdx.x * blockDim.x + threadIdx.x;
    if (idx >= E * H_) return;
    const int e = idx >> 3, h = idx & 7;
    const int s = src[e], d = dst[e];
    const float m = fdec_(segmax[d * H_ + h]);
    const float w = __expf(alpha[idx] - m);
    atomicAdd(&denom[d * H_ + h], w);
    const float* vv = Vsrc + ((size_t)s * H_ + h) * D_;
    const float* mr = m_rel + (size_t)h * D_ * D_;
    float me[D_];
#pragma unroll
    for (int f = 0; f < D_; ++f) me[f] = 0.f;
#pragma unroll
    for (int dd = 0; dd < D_; ++dd) {
        const float vd = vv[dd];
#pragma unroll
        for (int f = 0; f < D_; ++f) me[f] += vd * mr[dd * D_ + f];
    }
    float* np = num + ((size_t)d * H_ + h) * D_;
#pragma unroll
    for (int f = 0; f < D_; ++f) atomicAdd(&np[f], w * me[f]);
}

static inline dim3 gemm_grid(int M) { return dim3((((M + 15) / 16) + 7) / 8); }

extern "C" void kernel_launch(void* const* d_in, const int* in_sizes, int n_in,
                              void* d_out, int out_size, void* d_ws, size_t ws_size,
                              hipStream_t stream) {
    (void)in_sizes; (void)n_in; (void)out_size;
    const float* x_m    = (const float*)d_in[0];
    const float* x_d    = (const float*)d_in[1];
    const float* x_a    = (const float*)d_in[2];
    const int*   src_dm = (const int*)d_in[3];
    const int*   dst_dm = (const int*)d_in[4];
    const int*   src_am = (const int*)d_in[5];
    const int*   dst_am = (const int*)d_in[6];
    const float* Wpre_m = (const float*)d_in[11];
    const float* Wpre_d = (const float*)d_in[12];
    const float* Wpre_a = (const float*)d_in[13];
    const float* bpre   = (const float*)d_in[14];
    const float* Wk     = (const float*)d_in[15];
    const float* bk     = (const float*)d_in[16];
    const float* Wq     = (const float*)d_in[17];
    const float* bq     = (const float*)d_in[18];
    const float* Wv     = (const float*)d_in[19];
    const float* bv     = (const float*)d_in[20];
    const float* a_rel  = (const float*)d_in[21];
    const float* m_rel  = (const float*)d_in[22];
    const float* p_rel  = (const float*)d_in[23];
    const float* skip   = (const float*)d_in[24];
    const float* Wa     = (const float*)d_in[25];
    const float* ba     = (const float*)d_in[26];
    const float* Wlin   = (const float*)d_in[27];
    const float* blin   = (const float*)d_in[28];
    float* out = (float*)d_out;

    // workspace layout (floats)
    float* ws = (float*)d_ws;
    size_t o = 0;
    float* xs_m   = ws + o; o += (size_t)NM_ * HID_;
    float* xs_d   = ws + o; o += (size_t)ND_ * HID_;
    float* xs_a   = ws + o; o += (size_t)NA_ * HID_;
    float* q_m    = ws + o; o += (size_t)NM_ * HID_;
    float* k_d    = ws + o; o += (size_t)ND_ * HID_;
    float* v_d    = ws + o; o += (size_t)ND_ * HID_;
    float* k_a    = ws + o; o += (size_t)NA_ * HID_;
    float* v_a    = ws + o; o += (size_t)NA_ * HID_;
    float* alpha0 = ws + o; o += (size_t)E_ * H_;
    float* alpha1 = ws + o; o += (size_t)E_ * H_;
    unsigned* segmax = (unsigned*)(ws + o); o += (size_t)NM_ * H_;
    float* denom  = ws + o; o += (size_t)NM_ * H_;
    float* num    = ws + o; o += (size_t)NM_ * HID_;
    float* hg     = ws + o; o += (size_t)NM_ * HID_;
    float* wpad   = ws + o; o += (size_t)HID_ * 16;
    float* bpad   = ws + o; o += 16;
    if (ws_size < o * sizeof(float)) return;

    const unsigned smem128 = 128 * 128 * sizeof(float);  // 64 KB
    const unsigned smem16  = 128 * 16 * sizeof(float);   // 8 KB

    // re-init atomic accumulation buffers every call (graph-replay safe)
    {
        int n1 = NM_ * H_, n2 = NM_ * HID_;
        fill_u32_kernel<<<(n1 + 255) / 256, 256, 0, stream>>>(segmax, ENC_NEG_INF, n1);
        fill_u32_kernel<<<(n1 + 255) / 256, 256, 0, stream>>>((unsigned*)denom, 0u, n1);
        fill_u32_kernel<<<(n2 + 255) / 256, 256, 0, stream>>>((unsigned*)num, 0u, n2);
        pad_head_kernel<<<(HID_ * 16 + 255) / 256, 256, 0, stream>>>(Wlin, blin, wpad, bpad);
    }

    // pre-encoders: xs[t] = x_t @ Wpre_t + bpre[t]
    gemm_wmma_kernel<256, 8, true, false, false><<<gemm_grid(NM_), 256, smem128, stream>>>(
        x_m, Wpre_m, bpre + 0 * HID_, nullptr, nullptr, nullptr, xs_m, NM_, HID_, HID_);
    gemm_wmma_kernel<128, 8, true, false, false><<<gemm_grid(ND_), 256, smem128, stream>>>(
        x_d, Wpre_d, bpre + 1 * HID_, nullptr, nullptr, nullptr, xs_d, ND_, HID_, HID_);
    gemm_wmma_kernel<128, 8, true, false, false><<<gemm_grid(NA_), 256, smem128, stream>>>(
        x_a, Wpre_a, bpre + 2 * HID_, nullptr, nullptr, nullptr, xs_a, NA_, HID_, HID_);

    // only projections that reach the movie output: q_movie, k/v_director, k/v_actor
    gemm_wmma_kernel<128, 8, true, false, false><<<gemm_grid(NM_), 256, smem128, stream>>>(
        xs_m, Wq + 0 * HID_ * HID_, bq + 0 * HID_, nullptr, nullptr, nullptr, q_m, NM_, HID_, HID_);
    gemm_wmma_kernel<128, 8, true, false, false><<<gemm_grid(ND_), 256, smem128, stream>>>(
        xs_d, Wk + 1 * HID_ * HID_, bk + 1 * HID_, nullptr, nullptr, nullptr, k_d, ND_, HID_, HID_);
    gemm_wmma_kernel<128, 8, true, false, false><<<gemm_grid(ND_), 256, smem128, stream>>>(
        xs_d, Wv + 1 * HID_ * HID_, bv + 1 * HID_, nullptr, nullptr, nullptr, v_d, ND_, HID_, HID_);
    gemm_wmma_kernel<128, 8, true, false, false><<<gemm_grid(NA_), 256, smem128, stream>>>(
        xs_a, Wk + 2 * HID_ * HID_, bk + 2 * HID_, nullptr, nullptr, nullptr, k_a, NA_, HID_, HID_);
    gemm_wmma_kernel<128, 8, true, false, false><<<gemm_grid(NA_), 256, smem128, stream>>>(
        xs_a, Wv + 2 * HID_ * HID_, bv + 2 * HID_, nullptr, nullptr, nullptr, v_a, NA_, HID_, HID_);

    // edge attention into movie nodes: relation 0 (director->movie), 1 (actor->movie)
    const int eg = (E_ * H_ + 255) / 256;
    edge_alpha_kernel<<<eg, 256, 0, stream>>>(src_dm, dst_dm, k_d, q_m,
        a_rel + 0 * H_ * D_ * D_, p_rel + 0 * H_, alpha0, segmax, E_);
    edge_alpha_kernel<<<eg, 256, 0, stream>>>(src_am, dst_am, k_a, q_m,
        a_rel + 1 * H_ * D_ * D_, p_rel + 1 * H_, alpha1, segmax, E_);
    edge_accum_kernel<<<eg, 256, 0, stream>>>(src_dm, dst_dm, v_d,
        m_rel + 0 * H_ * D_ * D_, alpha0, segmax, denom, num, E_);
    edge_accum_kernel<<<eg, 256, 0, stream>>>(src_am, dst_am, v_a,
        m_rel + 1 * H_ * D_ * D_, alpha1, segmax, denom, num, E_);

    // hg = g * (gelu(num/denom) @ Wa[0] + ba[0]) + (1-g) * xs_m, g = sigmoid(skip[0])
    gemm_wmma_kernel<128, 8, true, true, true><<<gemm_grid(NM_), 256, smem128, stream>>>(
        num, Wa + 0 * HID_ * HID_, ba + 0 * HID_, denom, xs_m, skip + 0, hg, NM_, HID_, HID_);

    // out = hg @ Wlin + blin  (B zero-padded to 16 cols -> guard-free inner loop)
    gemm_wmma_kernel<128, 1, true, false, false><<<gemm_grid(NM_), 256, smem16, stream>>>(
        hg, wpad, bpad, nullptr, nullptr, nullptr, out, NM_, NCLS_, NCLS_);
}